// PPNetLayer_86955907875096
// MI455X (gfx1250) — compile-verified
//
#include <hip/hip_runtime.h>
#include <math.h>

// ---------------------------------------------------------------------------
// PPNet fused kernel for MI455X (gfx1250), wave32, WMMA f32 16x16x4.
// B=8192, P=256, G=512, T=8, H0=512, H1=256, C=P+G=768.
// One workgroup = 64 batch rows x 1 task; full 7-GEMM chain fused via LDS.
// Weights are repacked once per launch into [K/2][N][2] "B-fragment" order
// in d_ws so each WMMA B operand is a single global_load_b64, and each wave
// covers all 4 row-bands so one B fragment feeds 4 WMMAs.
// ---------------------------------------------------------------------------

typedef __attribute__((ext_vector_type(2))) float v2f;
typedef __attribute__((ext_vector_type(8))) float v8f;

#define B_   8192
#define P_   256
#define G_   512
#define T_   8
#define H0_  512
#define H1_  256
#define C_   768

#define MTILE 64          // batch rows per workgroup
#define KC    64          // K-chunk for global-activation staging
#define LDA0  516         // Act   row stride (512 + 4 pad; 516 % 64 == 4)
#define LDA1  260         // Act1  row stride (256 + 4 pad)
#define LDAS  68          // chunk row stride (64 + 4 pad)

static __device__ __forceinline__ v8f wmma4(v2f a, v2f b, v8f c) {
  // D = A(16x4) * B(4x16) + C, fp32, RNE.
  return __builtin_amdgcn_wmma_f32_16x16x4_f32(false, a, false, b,
                                               (short)0, c, false, false);
}

static __device__ __forceinline__ v8f vzero8() {
  v8f z = {0.f, 0.f, 0.f, 0.f, 0.f, 0.f, 0.f, 0.f};
  return z;
}

// WMMA GEMM over all 4 row-bands of the 64-row tile:
//   acc[band][tl] += A[band*16.., 0..K) * W[0..K, ncol0 + tl*16 ..)
// A: LDS row-major, stride lda. W: global, pre-offset to first K row.
// PACKED: W in [K/2][N][2] pair-interleaved order -> one b64 per fragment.
template <int NT, bool PACKED>
static __device__ __forceinline__ void gemm_bands(
    const float* __restrict__ A, int lda,
    const float* __restrict__ W, int ldw, int K,
    int ncol0, int lane, v8f (&acc)[4][NT]) {
  const int lm   = lane & 15;
  const int lh   = lane >> 4;     // 0/1 -> K sub-pair 0/1
  const int ksub = lh << 1;
  for (int k = 0; k < K; k += 4) {
    v2f a[4];
#pragma unroll
    for (int b = 0; b < 4; ++b) {
      const float2 av = *(const float2*)(A + (b * 16 + lm) * lda + k + ksub);
      a[b][0] = av.x; a[b][1] = av.y;
    }
    v2f bf[NT];
#pragma unroll
    for (int tl = 0; tl < NT; ++tl) {
      const int n = ncol0 + tl * 16 + lm;
      if (PACKED) {
        const float2 wv =
            *(const float2*)(W + ((size_t)((k >> 1) + lh) * ldw + n) * 2);
        bf[tl][0] = wv.x; bf[tl][1] = wv.y;
      } else {
        const float* wr = W + (size_t)(k + ksub) * ldw + n;
        bf[tl][0] = wr[0];
        bf[tl][1] = wr[ldw];
      }
    }
#pragma unroll
    for (int b = 0; b < 4; ++b)
#pragma unroll
      for (int tl = 0; tl < NT; ++tl)
        acc[b][tl] = wmma4(a[b], bf[tl], acc[b][tl]);
  }
}

// Stage a [64 x 64] chunk of comb = concat(prior, general) into LDS.
static __device__ __forceinline__ void load_chunk_comb(
    float* __restrict__ As, const float* __restrict__ prior,
    const float* __restrict__ general, int row0, int k0, int tid) {
  for (int i = tid; i < MTILE * KC; i += 256) {
    const int r = i >> 6, c = i & 63;
    const int k = k0 + c;
    const float v = (k < P_)
        ? prior[(size_t)(row0 + r) * P_ + k]
        : general[(size_t)(row0 + r) * G_ + (k - P_)];
    As[r * LDAS + c] = v;
  }
}

// Stage a [64 x 64] chunk of a plain [B x ldsrc] matrix into LDS.
static __device__ __forceinline__ void load_chunk(
    float* __restrict__ As, const float* __restrict__ src, int ldsrc,
    int row0, int k0, int tid) {
  for (int i = tid; i < MTILE * KC; i += 256) {
    const int r = i >> 6, c = i & 63;
    As[r * LDAS + c] = src[(size_t)(row0 + r) * ldsrc + k0 + c];
  }
}

// Repack W[T,K,N] -> Wp[T][K/2][N][2] (k-pairs innermost).
__global__ void repack_pairs_kernel(const float* __restrict__ src,
                                    float* __restrict__ dst,
                                    int K, int N, long total) {
  const long kn = (long)K * N;
  for (long i = (long)blockIdx.x * blockDim.x + threadIdx.x; i < total;
       i += (long)gridDim.x * blockDim.x) {
    const long t = i / kn;
    const long r = i - t * kn;
    const int  k = (int)(r / N);
    const int  n = (int)(r - (long)k * N);
    dst[t * kn + (long)(k >> 1) * 2 * N + 2 * n + (k & 1)] = src[i];
  }
}

template <bool PACKED>
__launch_bounds__(256)
__global__ void ppnet_fused_kernel(
    const float* __restrict__ prior,   // [B,P]
    const float* __restrict__ general, // [B,G]
    const float* __restrict__ pers,    // [B,G]
    const float* __restrict__ tW0,     // [T,G,H0]   (packed or raw)
    const float* __restrict__ tb0,     // [T,H0]
    const float* __restrict__ tW1,     // [T,H0,H1]  (packed or raw)
    const float* __restrict__ tb1,     // [T,H1]
    const float* __restrict__ tW2,     // [T,H1,1]
    const float* __restrict__ tb2,     // [T,1]
    const float* __restrict__ p0W,     // [T,C,H0]   (packed or raw)
    const float* __restrict__ p0b,     // [T,H0]
    const float* __restrict__ g0W,     // [T,H0,H0]  (packed or raw)
    const float* __restrict__ g0b,     // [T,H0]
    const float* __restrict__ p1W,     // [T,C,H1]   (packed or raw)
    const float* __restrict__ p1b,     // [T,H1]
    const float* __restrict__ g1W,     // [T,H1,H1]  (packed or raw)
    const float* __restrict__ g1b,     // [T,H1]
    float* __restrict__ out) {         // [B,T]
  __shared__ float Act [MTILE * LDA0]; // h0 -> g0 -> x0   (64x512)
  __shared__ float Act1[MTILE * LDA1]; // h1 -> g1 -> x1   (64x256)
  __shared__ float As  [MTILE * LDAS]; // K-chunk staging  (64x64)

  const int tid  = threadIdx.x;
  const int lane = tid & 31;
  const int wave = tid >> 5;
  const int row0 = blockIdx.x * MTILE;
  const int t    = blockIdx.y;

  const int lm = lane & 15;
  const int lh = lane >> 4;           // D rows +0 / +8 within a band
  const int nc512 = wave * 64;        // this wave's column group, N=512 stages
  const int nc256 = wave * 32;        // this wave's column group, N=256 stages

  const float* p0Wt = p0W + (size_t)t * C_  * H0_;
  const float* g0Wt = g0W + (size_t)t * H0_ * H0_;
  const float* tW0t = tW0 + (size_t)t * G_  * H0_;
  const float* p1Wt = p1W + (size_t)t * C_  * H1_;
  const float* g1Wt = g1W + (size_t)t * H1_ * H1_;
  const float* tW1t = tW1 + (size_t)t * H0_ * H1_;

  // ---- Stage 1: h0 = relu(comb @ p0W + b)  ->  Act -----------------------
  {
    v8f acc[4][4];
#pragma unroll
    for (int b = 0; b < 4; ++b)
#pragma unroll
      for (int i = 0; i < 4; ++i) acc[b][i] = vzero8();
    for (int kc = 0; kc < C_; kc += KC) {
      __syncthreads();
      load_chunk_comb(As, prior, general, row0, kc, tid);
      __syncthreads();
      gemm_bands<4, PACKED>(As, LDAS, p0Wt + (size_t)kc * H0_, H0_, KC,
                            nc512, lane, acc);
    }
    const float* bp = p0b + (size_t)t * H0_;
#pragma unroll
    for (int tl = 0; tl < 4; ++tl) {
      const int n = nc512 + tl * 16 + lm;
      const float bn = bp[n];
#pragma unroll
      for (int b = 0; b < 4; ++b) {
        const int mD = b * 16 + lh * 8;
#pragma unroll
        for (int v = 0; v < 8; ++v)
          Act[(mD + v) * LDA0 + n] = fmaxf(acc[b][tl][v] + bn, 0.f);
      }
    }
  }
  __syncthreads();

  // ---- Stage 2: g0 = h0 * sigmoid(2*(h0 @ g0W + b))  (in place) ----------
  {
    v8f acc[4][4];
#pragma unroll
    for (int b = 0; b < 4; ++b)
#pragma unroll
      for (int i = 0; i < 4; ++i) acc[b][i] = vzero8();
    gemm_bands<4, PACKED>(Act, LDA0, g0Wt, H0_, H0_, nc512, lane, acc);
    __syncthreads();  // all reads of h0 complete before overwrite
    const float* bp = g0b + (size_t)t * H0_;
#pragma unroll
    for (int tl = 0; tl < 4; ++tl) {
      const int n = nc512 + tl * 16 + lm;
      const float bn = bp[n];
#pragma unroll
      for (int b = 0; b < 4; ++b) {
        const int mD = b * 16 + lh * 8;
#pragma unroll
        for (int v = 0; v < 8; ++v) {
          const int idx = (mD + v) * LDA0 + n;
          const float z = 2.f * (acc[b][tl][v] + bn);
          Act[idx] = Act[idx] * (1.f / (1.f + __expf(-z)));
        }
      }
    }
  }

  // ---- Stage 3: x0 = relu((pers @ tW0 + b) * g0)  ->  Act (in place) -----
  {
    v8f acc[4][4];
#pragma unroll
    for (int b = 0; b < 4; ++b)
#pragma unroll
      for (int i = 0; i < 4; ++i) acc[b][i] = vzero8();
    for (int kc = 0; kc < G_; kc += KC) {
      __syncthreads();
      load_chunk(As, pers, G_, row0, kc, tid);
      __syncthreads();
      gemm_bands<4, PACKED>(As, LDAS, tW0t + (size_t)kc * H0_, H0_, KC,
                            nc512, lane, acc);
    }
    const float* bp = tb0 + (size_t)t * H0_;
#pragma unroll
    for (int tl = 0; tl < 4; ++tl) {
      const int n = nc512 + tl * 16 + lm;
      const float bn = bp[n];
#pragma unroll
      for (int b = 0; b < 4; ++b) {
        const int mD = b * 16 + lh * 8;
#pragma unroll
        for (int v = 0; v < 8; ++v) {
          const int idx = (mD + v) * LDA0 + n;
          Act[idx] = fmaxf((acc[b][tl][v] + bn) * Act[idx], 0.f);
        }
      }
    }
  }
  __syncthreads();

  // ---- Stage 4: h1 = relu(comb @ p1W + b)  ->  Act1 ----------------------
  {
    v8f acc[4][2];
#pragma unroll
    for (int b = 0; b < 4; ++b)
#pragma unroll
      for (int i = 0; i < 2; ++i) acc[b][i] = vzero8();
    for (int kc = 0; kc < C_; kc += KC) {
      __syncthreads();
      load_chunk_comb(As, prior, general, row0, kc, tid);
      __syncthreads();
      gemm_bands<2, PACKED>(As, LDAS, p1Wt + (size_t)kc * H1_, H1_, KC,
                            nc256, lane, acc);
    }
    const float* bp = p1b + (size_t)t * H1_;
#pragma unroll
    for (int tl = 0; tl < 2; ++tl) {
      const int n = nc256 + tl * 16 + lm;
      const float bn = bp[n];
#pragma unroll
      for (int b = 0; b < 4; ++b) {
        const int mD = b * 16 + lh * 8;
#pragma unroll
        for (int v = 0; v < 8; ++v)
          Act1[(mD + v) * LDA1 + n] = fmaxf(acc[b][tl][v] + bn, 0.f);
      }
    }
  }
  __syncthreads();

  // ---- Stage 5: g1 = h1 * sigmoid(2*(h1 @ g1W + b))  (in place) ----------
  {
    v8f acc[4][2];
#pragma unroll
    for (int b = 0; b < 4; ++b)
#pragma unroll
      for (int i = 0; i < 2; ++i) acc[b][i] = vzero8();
    gemm_bands<2, PACKED>(Act1, LDA1, g1Wt, H1_, H1_, nc256, lane, acc);
    __syncthreads();  // all reads of h1 complete before overwrite
    const float* bp = g1b + (size_t)t * H1_;
#pragma unroll
    for (int tl = 0; tl < 2; ++tl) {
      const int n = nc256 + tl * 16 + lm;
      const float bn = bp[n];
#pragma unroll
      for (int b = 0; b < 4; ++b) {
        const int mD = b * 16 + lh * 8;
#pragma unroll
        for (int v = 0; v < 8; ++v) {
          const int idx = (mD + v) * LDA1 + n;
          const float z = 2.f * (acc[b][tl][v] + bn);
          Act1[idx] = Act1[idx] * (1.f / (1.f + __expf(-z)));
        }
      }
    }
  }
  __syncthreads();

  // ---- Stage 6: x1 = relu((x0 @ tW1 + b) * g1)  ->  Act1 (in place) ------
  {
    v8f acc[4][2];
#pragma unroll
    for (int b = 0; b < 4; ++b)
#pragma unroll
      for (int i = 0; i < 2; ++i) acc[b][i] = vzero8();
    gemm_bands<2, PACKED>(Act, LDA0, tW1t, H1_, H0_, nc256, lane, acc);
    const float* bp = tb1 + (size_t)t * H1_;
#pragma unroll
    for (int tl = 0; tl < 2; ++tl) {
      const int n = nc256 + tl * 16 + lm;
      const float bn = bp[n];
#pragma unroll
      for (int b = 0; b < 4; ++b) {
        const int mD = b * 16 + lh * 8;
#pragma unroll
        for (int v = 0; v < 8; ++v) {
          const int idx = (mD + v) * LDA1 + n;
          Act1[idx] = fmaxf((acc[b][tl][v] + bn) * Act1[idx], 0.f);
        }
      }
    }
  }
  __syncthreads();

  // ---- Stage 7: out[:, t] = x1 @ tW2 + b2 --------------------------------
  if (tid < MTILE) {
    const float* w2 = tW2 + (size_t)t * H1_;
    float s = tb2[t];
#pragma unroll 4
    for (int n = 0; n < H1_; ++n) s += Act1[tid * LDA1 + n] * w2[n];
    out[(size_t)(row0 + tid) * T_ + t] = s;
  }
}

extern "C" void kernel_launch(void* const* d_in, const int* in_sizes, int n_in,
                              void* d_out, int out_size, void* d_ws,
                              size_t ws_size, hipStream_t stream) {
  (void)in_sizes; (void)n_in; (void)out_size;
  const float* prior   = (const float*)d_in[0];
  const float* general = (const float*)d_in[1];
  const float* pers    = (const float*)d_in[2];
  const float* tW0     = (const float*)d_in[3];
  const float* tb0     = (const float*)d_in[4];
  const float* tW1     = (const float*)d_in[5];
  const float* tb1     = (const float*)d_in[6];
  const float* tW2     = (const float*)d_in[7];
  const float* tb2     = (const float*)d_in[8];
  const float* p0W     = (const float*)d_in[9];
  const float* p0b     = (const float*)d_in[10];
  const float* g0W     = (const float*)d_in[11];
  const float* g0b     = (const float*)d_in[12];
  const float* p1W     = (const float*)d_in[13];
  const float* p1b     = (const float*)d_in[14];
  const float* g1W     = (const float*)d_in[15];
  const float* g1b     = (const float*)d_in[16];
  float* out = (float*)d_out;

  dim3 grid(B_ / MTILE, T_);  // 128 x 8 = 1024 workgroups
  dim3 block(256);            // 8 wave32s

  // Packed-weight workspace layout (floats):
  const long szP0 = (long)T_ * C_  * H0_;  // 3,145,728
  const long szG0 = (long)T_ * H0_ * H0_;  // 2,097,152
  const long szT0 = (long)T_ * G_  * H0_;  // 2,097,152
  const long szP1 = (long)T_ * C_  * H1_;  // 1,572,864
  const long szG1 = (long)T_ * H1_ * H1_;  //   524,288
  const long szT1 = (long)T_ * H0_ * H1_;  // 1,048,576
  const long totF = szP0 + szG0 + szT0 + szP1 + szG1 + szT1;  // 10,485,760

  if (ws_size >= (size_t)totF * sizeof(float)) {
    float* ws = (float*)d_ws;
    float* p0Wp = ws;
    float* g0Wp = p0Wp + szP0;
    float* tW0p = g0Wp + szG0;
    float* p1Wp = tW0p + szT0;
    float* g1Wp = p1Wp + szP1;
    float* tW1p = g1Wp + szG1;

    dim3 rg(2048), rb(256);
    repack_pairs_kernel<<<rg, rb, 0, stream>>>(p0W, p0Wp, C_,  H0_, szP0);
    repack_pairs_kernel<<<rg, rb, 0, stream>>>(g0W, g0Wp, H0_, H0_, szG0);
    repack_pairs_kernel<<<rg, rb, 0, stream>>>(tW0, tW0p, G_,  H0_, szT0);
    repack_pairs_kernel<<<rg, rb, 0, stream>>>(p1W, p1Wp, C_,  H1_, szP1);
    repack_pairs_kernel<<<rg, rb, 0, stream>>>(g1W, g1Wp, H1_, H1_, szG1);
    repack_pairs_kernel<<<rg, rb, 0, stream>>>(tW1, tW1p, H0_, H1_, szT1);

    ppnet_fused_kernel<true><<<grid, block, 0, stream>>>(
        prior, general, pers, tW0p, tb0, tW1p, tb1, tW2, tb2,
        p0Wp, p0b, g0Wp, g0b, p1Wp, p1b, g1Wp, g1b, out);
  } else {
    ppnet_fused_kernel<false><<<grid, block, 0, stream>>>(
        prior, general, pers, tW0, tb0, tW1, tb1, tW2, tb2,
        p0W, p0b, g0W, g0b, p1W, p1b, g1W, g1b, out);
  }
}